// Preprocessing_38079180046605
// MI455X (gfx1250) — compile-verified
//
#include <hip/hip_runtime.h>

// CDNA5 / gfx1250 wave32
typedef __attribute__((ext_vector_type(2))) float v2f;
typedef __attribute__((ext_vector_type(8))) float v8f;

#define NPTS   128   // points per trajectory
#define NALIGN 32    // align subset size (== wave size)
#define NTRAIN 64    // train subset size
#define BLOCK  256   // 8 waves per block, one batch per thread

__global__ __launch_bounds__(BLOCK)
void kabsch_align_kernel(const float* __restrict__ traj,      // [B,128,3]
                         const float* __restrict__ ref,       // [32,3] (pre-centered)
                         const int*   __restrict__ align_idx, // [32]
                         const int*   __restrict__ train_idx, // [64]
                         float*       __restrict__ out)       // [B,192]
{
    __shared__ int   s_train[NTRAIN];
    __shared__ float s_mt[BLOCK * 12];      // per-lane {M row-major (9), t (3)}
    __shared__ float s_out[BLOCK / 32][NTRAIN * 3];

    const int tid  = threadIdx.x;
    const int lane = tid & 31;
    const int wv   = tid >> 5;

    if (tid < NTRAIN) s_train[tid] = train_idx[tid];
    __syncthreads();

    // ---------- Phase 1: per-lane covariance + mean (one batch per lane) ----------
    const size_t b = (size_t)blockIdx.x * BLOCK + tid;
    const float* trow = traj + b * (NPTS * 3);

    float s00=0.f,s01=0.f,s02=0.f, s10=0.f,s11=0.f,s12=0.f, s20=0.f,s21=0.f,s22=0.f;
    float c0=0.f, c1=0.f, c2=0.f;
    for (int a = 0; a < NALIGN; ++a) {
        unsigned ia = (unsigned)align_idx[a];       // wave-uniform -> scalar load
        const float* p = trow + ia * 3u;            // per-lane gather (12 B)
        float px = p[0], py = p[1], pz = p[2];
        float rx = ref[a*3+0], ry = ref[a*3+1], rz = ref[a*3+2];  // uniform -> SMEM
        c0 += px; c1 += py; c2 += pz;
        s00 = fmaf(px,rx,s00); s01 = fmaf(px,ry,s01); s02 = fmaf(px,rz,s02);
        s10 = fmaf(py,rx,s10); s11 = fmaf(py,ry,s11); s12 = fmaf(py,rz,s12);
        s20 = fmaf(pz,rx,s20); s21 = fmaf(pz,ry,s21); s22 = fmaf(pz,rz,s22);
    }
    c0 *= (1.f/NALIGN); c1 *= (1.f/NALIGN); c2 *= (1.f/NALIGN);
    // (ref is centered, so S = sel^T ref == x0^T ref in exact arithmetic)

    // ---------- Phase 2: Horn quaternion (max eigvec of 4x4 N) per lane ----------
    float n00 =  s00 + s11 + s22;
    float n01 =  s12 - s21,  n02 = s20 - s02,  n03 = s01 - s10;
    float n11 =  s00 - s11 - s22, n12 = s01 + s10, n13 = s20 + s02;
    float n22 = -s00 + s11 - s22, n23 = s12 + s21;
    float n33 = -s00 - s11 + s22;

    float fro = sqrtf(n00*n00 + n11*n11 + n22*n22 + n33*n33 +
                      2.f*(n01*n01 + n02*n02 + n03*n03 + n12*n12 + n13*n13 + n23*n23))
                + 1e-12f;                       // shift: N + fro*I is PSD, keeps top eigvec

    float qw = 1.f, qx = 0.11f, qy = 0.22f, qz = 0.33f;
    #pragma unroll 4
    for (int it = 0; it < 24; ++it) {
        float tw = fmaf(n00,qw, fmaf(n01,qx, fmaf(n02,qy, fmaf(n03,qz, fro*qw))));
        float tx = fmaf(n01,qw, fmaf(n11,qx, fmaf(n12,qy, fmaf(n13,qz, fro*qx))));
        float ty = fmaf(n02,qw, fmaf(n12,qx, fmaf(n22,qy, fmaf(n23,qz, fro*qy))));
        float tz = fmaf(n03,qw, fmaf(n13,qx, fmaf(n23,qy, fmaf(n33,qz, fro*qz))));
        float rn = rsqrtf(fmaf(tw,tw, fmaf(tx,tx, fmaf(ty,ty, tz*tz))) + 1e-30f);
        qw = tw*rn; qx = tx*rn; qy = ty*rn; qz = tz*rn;
    }

    // R(q) rotates align-points onto ref; reference applies row-vec @ rotate => M = R^T
    float ww=qw*qw, xx=qx*qx, yy=qy*qy, zz=qz*qz;
    float R00=ww+xx-yy-zz, R01=2.f*(qx*qy - qw*qz), R02=2.f*(qx*qz + qw*qy);
    float R10=2.f*(qx*qy + qw*qz), R11=ww-xx+yy-zz, R12=2.f*(qy*qz - qw*qx);
    float R20=2.f*(qx*qz - qw*qy), R21=2.f*(qy*qz + qw*qx), R22=ww-xx-yy+zz;

    float M00=R00, M01=R10, M02=R20;
    float M10=R01, M11=R11, M12=R21;
    float M20=R02, M21=R12, M22=R22;
    // fold centering: out = p@M + t,  t = -(c@M)
    float t0 = -(c0*M00 + c1*M10 + c2*M20);
    float t1 = -(c0*M01 + c1*M11 + c2*M21);
    float t2 = -(c0*M02 + c1*M12 + c2*M22);

    float* slot = &s_mt[tid * 12];
    slot[0]=M00; slot[1]=M01; slot[2]=M02;
    slot[3]=M10; slot[4]=M11; slot[5]=M12;
    slot[6]=M20; slot[7]=M21; slot[8]=M22;
    slot[9]=t0;  slot[10]=t1; slot[11]=t2;
    asm volatile("s_wait_dscnt 0" ::: "memory");   // intra-wave LDS fence (HW is in-order)

    // ---------- Phase 3: apply via V_WMMA_F32_16X16X4_F32, one batch per iteration ----------
    // A (16x4): 16 train points [x,y,z,1];  B (4x16): [M; t] in columns 0..2;  D = A*B.
    // f32 A layout: lanes 0-15 = rows, VGPR0=K0,VGPR1=K1; lanes 16-31 VGPR0=K2,VGPR1=K3.
    const int  m16    = lane & 15;
    const int  half   = lane >> 4;
    const bool valid3 = (m16 < 3);
    float* wout = s_out[wv];

    // Loop-invariant per-lane A-gather offsets (uint32 -> SADDR+voffset form).
    // One 8-byte gather per WMMA: half0 lanes need {x,y} at pt*3, half1 lanes
    // need {z, (ignored)} at pt*3+2 -- K3 of A is the constant 1.0.
    unsigned off8[4];
    #pragma unroll
    for (int g = 0; g < 4; ++g) {
        unsigned pt = (unsigned)s_train[g * 16 + m16];
        off8[g] = pt * 3u + (unsigned)(half * 2);
    }
    // Loop-invariant in-range B-operand LDS offsets (clamped; zeroed by select below)
    const int bidx  = m16 & 3;
    const int boff0 = (half ? 6 : 0) + bidx;   // M[2][n] : M[0][n]
    const int boff1 = (half ? 9 : 3) + bidx;   // t[n]    : M[1][n]

    for (int r = 0; r < 32; ++r) {
        const size_t br = (size_t)blockIdx.x * BLOCK + wv * 32 + r;
        const float* mt = &s_mt[(wv * 32 + r) * 12];

        // Build B operand: unconditional loads + cndmask (no exec branches)
        float braw0 = mt[boff0];
        float braw1 = mt[boff1];
        v2f Bv;
        Bv.x = valid3 ? braw0 : 0.f;
        Bv.y = valid3 ? braw1 : 0.f;

        const float* tr = traj + br * (NPTS * 3);
        #pragma unroll
        for (int g = 0; g < 4; ++g) {
            struct { float lo, hi; } pr;
            __builtin_memcpy(&pr, tr + off8[g], 8);   // single b64 gather (4B-aligned ok)
            v2f Av;
            Av.x = pr.lo;                              // K0 (x) or K2 (z)
            Av.y = half ? 1.0f : pr.hi;                // K1 (y) or K3 (const 1)
            v8f C = {};
            v8f D = __builtin_amdgcn_wmma_f32_16x16x4_f32(
                        false, Av, false, Bv, (short)0, C, false, false);
            // D layout: VGPRr -> row r (lanes0-15) / row r+8 (lanes16-31), col = lane&15
            if (valid3) {
                const int baserow = g * 16 + (half ? 8 : 0);
                #pragma unroll
                for (int rr = 0; rr < 8; ++rr)
                    wout[(baserow + rr) * 3 + m16] = D[rr];
            }
        }
        asm volatile("s_wait_dscnt 0" ::: "memory");
        // coalesced writeback: 192 floats = 6 x (32 lanes x 4B contiguous)
        float* ob = out + br * (NTRAIN * 3);
        #pragma unroll
        for (int u = 0; u < 6; ++u)
            ob[u * 32 + lane] = wout[u * 32 + lane];
        asm volatile("" ::: "memory");
    }
}

extern "C" void kernel_launch(void* const* d_in, const int* in_sizes, int n_in,
                              void* d_out, int out_size, void* d_ws, size_t ws_size,
                              hipStream_t stream) {
    const float* traj      = (const float*)d_in[0];
    const float* ref_x     = (const float*)d_in[1];
    const int*   align_idx = (const int*)d_in[2];
    const int*   train_idx = (const int*)d_in[3];
    float*       out       = (float*)d_out;

    const int B = in_sizes[0] / (NPTS * 3);      // 65536
    dim3 grid(B / BLOCK), block(BLOCK);
    kabsch_align_kernel<<<grid, block, 0, stream>>>(traj, ref_x, align_idx, train_idx, out);
}